// HiePoolingLayer_27281632264964
// MI455X (gfx1250) — compile-verified
//
#include <hip/hip_runtime.h>

// HiePoolingLayer on MI455X (gfx1250, wave32).
// Bandwidth-bound: 161 MB traffic -> ~7us floor @ 23.3 TB/s. X is read exactly
// once. The per-16-segment mean pool is a banded f32 GEMM on the matrix pipe
// (V_WMMA_F32_16X16X4_F32) with in-register weight synthesis (w = 1/count
// inside the segment band, 0 outside). Round 2: software-pipelined B loads
// (load t+4 before WMMA on t) and a uniform-branch store epilogue with
// immediate-offset b32 stores (no per-store exec juggling).

typedef __attribute__((ext_vector_type(2))) float v2f;
typedef __attribute__((ext_vector_type(8))) float v8f;

#define B_ 32
#define T_ 4096
#define E_ 256

// ---------------------------------------------------------------------------
// Kernel 1: one wave per row. Rank every cut via ballot/popc prefix, record
// cut token positions + per-row cut count. Also emits the float n_cuts output.
// ---------------------------------------------------------------------------
__global__ void HiePool_scan_cuts(const int* __restrict__ batch_x,
                                  int* __restrict__ cutpos,
                                  int* __restrict__ ncuts,
                                  float* __restrict__ out_ncuts,
                                  int maxlen) {
    const int b    = blockIdx.x;
    const int lane = threadIdx.x;                 // 32 threads = 1 wave
    const int* row = batch_x + (size_t)b * T_;

    int running = 0;
    for (int t0 = 0; t0 < T_; t0 += 32) {
        const int flag = row[t0 + lane];
        const unsigned mask = (unsigned)__ballot(flag != 0);   // wave32 mask
        const int rank = running + __popc(mask & ((1u << lane) - 1u));
        if (flag != 0 && rank < maxlen)
            cutpos[(size_t)b * maxlen + rank] = t0 + lane;
        running += __popc(mask);
    }
    if (lane == 0) {
        ncuts[b]     = running;
        out_ncuts[b] = (float)running;            // n_cuts output as float
    }
}

// ---------------------------------------------------------------------------
// Kernel 2: 16 waves / block; block = (row b, 16-segment group k0), wave =
// 16-feature column group. Each wave builds D(16x16) = W(16xK) x X(Kx16) with
// V_WMMA_F32_16X16X4_F32, K swept over the contiguous token band of the 16
// segments in steps of 4, software-pipelined one chunk ahead.
//
// ISA layouts (CDNA5 ISA):
//   A 16x4 f32 : lanes 0-15 -> M=lane, VGPR0/1 = K0/K1; lanes 16-31 -> K2/K3
//   B 4x16 f32 : row-striped across lanes; lanes 0-15 -> K0/K1, 16-31 -> K2/K3
//   C/D 16x16  : VGPR j -> M=j (lanes 0-15) / M=j+8 (lanes 16-31), N = lane&15
// ---------------------------------------------------------------------------
__global__ void HiePool_wmma(const float* __restrict__ X,
                             const int* __restrict__ cutpos,
                             const int* __restrict__ ncuts,
                             float* __restrict__ out,
                             int maxlen, int segGroups) {
    const int tid   = threadIdx.x;
    const int lane  = tid & 31;
    const int wave  = tid >> 5;                   // 0..15 -> feature group
    const int b     = blockIdx.x / segGroups;
    const int k0    = (blockIdx.x % segGroups) * 16;
    const int feat0 = wave * 16;

    const int nc = ncuts[b];
    const int m  = lane & 15;                     // segment row M
    const int lh = lane >> 4;                     // lane-half: K pair select
    const int n  = lane & 15;                     // feature column N

    const int* cp = cutpos + (size_t)b * maxlen;

    // Per-lane segment band (lo, hi] and weight 1/count; pad segments -> 0.
    int   lo = -2, hi = -2;
    float w  = 0.0f;
    const int seg = k0 + m;
    if (seg < nc) {
        hi = cp[seg];
        lo = (seg == 0) ? -1 : cp[seg - 1];
        w  = 1.0f / (float)(hi - lo);
    }

    v8f c = {0.f, 0.f, 0.f, 0.f, 0.f, 0.f, 0.f, 0.f};

    // Uniform token span covered by this 16-segment group.
    const int klast = min(k0 + 15, nc - 1);
    if (klast >= k0) {
        const int tstart = (k0 == 0) ? 0 : (cp[k0 - 1] + 1);
        const int tend   = cp[klast];
        const float* Xcol = X + (size_t)b * T_ * E_ + feat0 + n;

        // B-chunk loader (clamped at T-1; weights are 0 past the band).
        auto loadB = [&](int t) -> v2f {
            const int ta = t + 2 * lh;
            const int ca = (ta     < T_) ? ta     : (T_ - 1);
            const int cb = (ta + 1 < T_) ? ta + 1 : (T_ - 1);
            v2f r;
            r.x = Xcol[(size_t)ca * E_];
            r.y = Xcol[(size_t)cb * E_];
            return r;
        };
        // A-chunk weights for tokens [t, t+4).
        auto weights = [&](int t) -> v2f {
            const int ta = t + 2 * lh;
            v2f a;
            a.x = (ta     > lo && ta     <= hi) ? w : 0.0f;
            a.y = (ta + 1 > lo && ta + 1 <= hi) ? w : 0.0f;
            return a;
        };

        int t   = tstart;
        v2f cur = loadB(t);                       // pipeline preload
        for (; t + 4 <= tend; t += 4) {
            v2f nxt = loadB(t + 4);               // issue next chunk's loads
            // global_prefetch_b8 ~16 tokens ahead (clamped: no divergence).
            const int pf = (t + 16 < T_) ? (t + 16) : (T_ - 1);
            __builtin_prefetch(&Xcol[(size_t)pf * E_], 0, 1);
            c = __builtin_amdgcn_wmma_f32_16x16x4_f32(
                    false, weights(t), false, cur, (short)0, c, false, false);
            cur = nxt;
        }
        c = __builtin_amdgcn_wmma_f32_16x16x4_f32(
                false, weights(t), false, cur, (short)0, c, false, false);
    }

    // D store: per-lane base pointer, 8 b32 stores with immediate offsets.
    float* outp = out + ((size_t)b * maxlen + k0 + 8 * lh) * E_ + feat0 + n;
    if (k0 + 16 <= maxlen) {                      // uniform: full tile
#pragma unroll
        for (int j = 0; j < 8; ++j)
            outp[(size_t)j * E_] = c[j];
    } else {                                      // ragged tail tile only
#pragma unroll
        for (int j = 0; j < 8; ++j)
            if (k0 + j + 8 * lh < maxlen)
                outp[(size_t)j * E_] = c[j];
    }
}

// ---------------------------------------------------------------------------
extern "C" void kernel_launch(void* const* d_in, const int* in_sizes, int n_in,
                              void* d_out, int out_size, void* d_ws, size_t ws_size,
                              hipStream_t stream) {
    const float* X       = (const float*)d_in[0];   // nn_outs (B,T,E) f32
    const int*   batch_x = (const int*)d_in[1];     // cut markers (B,T)
    float*       out     = (float*)d_out;

    int maxlen = (out_size - B_) / (B_ * E_);       // padded segment count
    if (maxlen <= 0) maxlen = T_ / 16;

    int* cutpos = (int*)d_ws;                       // B * maxlen ints
    int* ncutsp = cutpos + (size_t)B_ * maxlen;     // B ints
    float* out_nc = out + (size_t)B_ * maxlen * E_; // n_cuts tail

    HiePool_scan_cuts<<<B_, 32, 0, stream>>>(batch_x, cutpos, ncutsp,
                                             out_nc, maxlen);

    const int segGroups = (maxlen + 15) / 16;
    HiePool_wmma<<<B_ * segGroups, 16 * 32, 0, stream>>>(X, cutpos, ncutsp,
                                                         out, maxlen, segGroups);
}